// kernel_generated_0_21887153341348
// MI455X (gfx1250) — compile-verified
//
#include <hip/hip_runtime.h>
#include <stdint.h>

typedef __attribute__((ext_vector_type(2))) float v2f;
typedef __attribute__((ext_vector_type(8))) float v8f;
typedef __attribute__((ext_vector_type(4))) unsigned int v4u;
typedef __attribute__((ext_vector_type(8))) unsigned int v8u;

#define Hh   56
#define Ww   56
#define CIN  128
#define NP   4      // p (w1 rows)
#define NJ   2      // group channel
#define KDIM 18     // (j,i,k) contraction, padded to 20 for 5 x K4 WMMA steps
#define NCOL (NP*Ww)     // 224 GEMM columns = (p,n)
#define NTILES (NCOL/16) // 14

// Dynamic-LDS arena layout (dynamic LDS starts at offset 0: no static LDS)
#define XS_ELEMS (CIN*Ww)            // 7168 f32 : x[:, o', :]
#define T4_OFF   XS_ELEMS            // 448 f32  : t4[p,j,n]
#define W1_OFF   (T4_OFF + NP*NJ*Ww) // 256 f32  : w1
#define SMEM_ELEMS (W1_OFF + NP*64)  // 7872 f32 = 31488 B

__global__ __launch_bounds__(128)
void fused_unfold_wmma(const float* __restrict__ x,
                       const float* __restrict__ w0,   // [32][2][3][3] == [32][18]
                       const float* __restrict__ w1,   // [4][64]
                       float* __restrict__ out) {
  extern __shared__ float smem[];
  float* xs  = smem;            // LDS byte offset 0 (TDM destination)
  float* t4s = smem + T4_OFF;
  float* ws1 = smem + W1_OFF;

  const int o    = blockIdx.x;                 // output row (after roll)
  const int op   = (o + Hh - 1) % Hh;          // source row
  const int tid  = threadIdx.x;
  const int lane = tid & 31;                   // wave32
  const int wave = tid >> 5;                   // 4 waves

  // ---- Phase 0: TDM DMA of the 2D tile x[:, o', :] (128 x 56 f32) to LDS --
  if (wave == 0) {
    uint64_t gaddr = (uint64_t)(uintptr_t)x + (uint64_t)op * (Ww * sizeof(float));
    v4u g0;
    g0.x = 1u;                                        // count=1, user descriptor
    g0.y = 0u;                                        // lds_addr = 0 (xs)
    g0.z = (uint32_t)gaddr;                           // global_addr[31:0]
    g0.w = (uint32_t)((gaddr >> 32) & 0x1FFFFFFull)   // global_addr[56:32]
         | (2u << 30);                                // type = 2 ("image")
    v8u g1;
    g1[0] = 2u << 16;                                 // data_size = 4 bytes
    g1[1] = (uint32_t)(Hh * Ww) << 16;                // tensor_dim0 = 3136
    g1[2] = (uint32_t)CIN << 16;                      // tensor_dim1 = 128
    g1[3] = (uint32_t)Ww << 16;                       // tile_dim0  = 56
    g1[4] = (uint32_t)CIN;                            // tile_dim1  = 128
    g1[5] = (uint32_t)(Hh * Ww);                      // tensor_dim0_stride = 3136
    g1[6] = ((uint32_t)(CIN * Hh * Ww) & 0xFFFFu) << 16; // dim1_stride[15:0]
    g1[7] = (uint32_t)(CIN * Hh * Ww) >> 16;             // dim1_stride[47:16]
    asm volatile("tensor_load_to_lds %0, %1" :: "s"(g0), "s"(g1) : "memory");
  }

  // Overlap with the DMA: stage w1 (1 KB) into LDS
  for (int f = tid; f < NP * 64; f += 128) ws1[f] = w1[f];

  if (wave == 0) __builtin_amdgcn_s_wait_tensorcnt(0);
  __syncthreads();   // publish TDM tile + ws1 to all waves

  // ---- Phase 1: t4[p,j,n] = sum_q w1[p,q] * xs[2q+j, n] -------------------
  for (int idx = tid; idx < NP * NJ * Ww; idx += 128) {
    int pj = idx / Ww, n = idx - pj * Ww;
    int p = pj >> 1, j = pj & 1;
    float acc = 0.f;
#pragma unroll 8
    for (int q = 0; q < 64; ++q)
      acc = __builtin_fmaf(ws1[p * 64 + q], xs[(2 * q + j) * Ww + n], acc);
    t4s[idx] = acc;
  }
  __syncthreads();

  // ---- Phase 2: out-row GEMM  M=32(l) x K=18(j,i,k) x N=224(p,n) ----------
  // A operand: w0 as [32,18], zero-padded to K=20. V_WMMA_F32_16X16X4_F32
  // layout: lanes 0-15 -> K={0,1}, lanes 16-31 -> K={2,3}, M = lane&15.
  const int koff = (lane < 16) ? 0 : 2;
  v2f amat[2][5];
#pragma unroll
  for (int mt = 0; mt < 2; ++mt) {
    int m = mt * 16 + (lane & 15);
#pragma unroll
    for (int kk = 0; kk < 5; ++kk) {
      int K0 = kk * 4 + koff;
      v2f a;
      a.x = (K0     < KDIM) ? w0[m * KDIM + K0    ] : 0.f;
      a.y = (K0 + 1 < KDIM) ? w0[m * KDIM + K0 + 1] : 0.f;
      amat[mt][kk] = a;
    }
  }

  for (int nt = wave; nt < NTILES; nt += 4) {
    const int c = nt * 16 + (lane & 15);   // GEMM column
    const int p = c / Ww, n = c - p * Ww;

    // B operand (4x16 per K-step): row = K, col = lane&15; element is the
    // double-unfolded, double-masked t4 value.
    v2f bmat[5];
#pragma unroll
    for (int kk = 0; kk < 5; ++kk) {
      float bv[2];
#pragma unroll
      for (int h = 0; h < 2; ++h) {
        int K = kk * 4 + koff + h;
        float val = 0.f;
        if (K < KDIM) {
          int j = K / 9, r = K - j * 9;
          int i = r / 3, k = r - i * 3;
          int u = n + k - 1;          // first-unfold pad check
          int v = u + i - 1;          // second-unfold pad check + src col
          if (u >= 0 && u < Ww && v >= 0 && v < Ww)
            val = t4s[(p * NJ + j) * Ww + v];
        }
        bv[h] = val;
      }
      v2f b; b.x = bv[0]; b.y = bv[1];
      bmat[kk] = b;
    }

#pragma unroll
    for (int mt = 0; mt < 2; ++mt) {
      v8f acc = {};
#pragma unroll
      for (int kk = 0; kk < 5; ++kk)
        acc = __builtin_amdgcn_wmma_f32_16x16x4_f32(
            false, amat[mt][kk], false, bmat[kk],
            (short)0, acc, false, false);

      // D layout: VGPR v -> row M = v + (lane<16 ? 0 : 8), col = lane&15.
      const int row_add = (lane < 16) ? 0 : 8;
#pragma unroll
      for (int v = 0; v < 8; ++v) {
        int l  = mt * 16 + v + row_add;
        int ch = l * 4 + p;                     // reshape [32,4] -> 128 chans
        out[ch * (Hh * Ww) + o * Ww + n] = acc[v];
      }
    }
  }
}

extern "C" void kernel_launch(void* const* d_in, const int* in_sizes, int n_in,
                              void* d_out, int out_size, void* d_ws, size_t ws_size,
                              hipStream_t stream) {
  const float* x  = (const float*)d_in[0];   // [1,128,56,56]
  const float* w0 = (const float*)d_in[1];   // [32,2,3,3]
  const float* w1 = (const float*)d_in[2];   // [4,64]
  float* out = (float*)d_out;                // [1,128,56,56]
  fused_unfold_wmma<<<Hh, 128, SMEM_ELEMS * sizeof(float), stream>>>(x, w0, w1, out);
}